// MolecularGCN_19885698580985
// MI455X (gfx1250) — compile-verified
//
#include <hip/hip_runtime.h>

// ---- problem constants (match reference) ----
#define F_IN  128
#define F_OUT 64

typedef float v2f __attribute__((ext_vector_type(2)));
typedef float v8f __attribute__((ext_vector_type(8)));

__device__ __forceinline__ void atomAddF(float* p, float v) {
  // hardware global_atomic_add_f32 (non-returning)
  unsafeAtomicAdd(p, v);
}

// ------------------------------------------------------------------
// init: zero agg, deg=1 (self loop), zero pooled sums/counts
// ------------------------------------------------------------------
__global__ void k_init(float* __restrict__ agg, float* __restrict__ deg,
                       float* __restrict__ sums, float* __restrict__ counts,
                       int N, int G) {
  long long t = (long long)blockIdx.x * blockDim.x + threadIdx.x;
  long long total = (long long)N * F_OUT;
  if (t < total) agg[t] = 0.0f;
  if (t < N) deg[t] = 1.0f;                    // self-loop contribution
  if (t < (long long)G * F_OUT) sums[t] = 0.0f;
  if (t < G) counts[t] = 0.0f;
}

// ------------------------------------------------------------------
// degree over targets (col), on top of the self-loop "1"
// ------------------------------------------------------------------
__global__ void k_degree(const int* __restrict__ col, float* __restrict__ deg, int E) {
  int t = blockIdx.x * blockDim.x + threadIdx.x;
  if (t < E) atomAddF(&deg[col[t]], 1.0f);
}

// deg -> deg^{-1/2} in place (deg >= 1 always)
__global__ void k_rsqrt(float* __restrict__ deg, int N) {
  int t = blockIdx.x * blockDim.x + threadIdx.x;
  if (t < N) deg[t] = rsqrtf(deg[t]);
}

// ------------------------------------------------------------------
// x = node_feat @ W^T via V_WMMA_F32_16X16X4_F32
// block = 128 threads = 4 waves; block b -> row tile b; wave w -> col tile w
// N must be a multiple of 16 (100000 = 6250*16), F_OUT = 4*16.
// ------------------------------------------------------------------
__global__ void __launch_bounds__(128)
k_gemm_wmma(const float* __restrict__ X, const float* __restrict__ W,
            float* __restrict__ Y) {
  const int wave = threadIdx.x >> 5;       // 0..3 -> output col tile
  const int lane = threadIdx.x & 31;
  const int m    = lane & 15;
  const int half = lane >> 4;              // 0: K pair {0,1}; 1: K pair {2,3}

  const float* arow = X + (long long)(blockIdx.x * 16 + m) * F_IN; // A row (node)
  const float* brow = W + (long long)(wave * 16 + m) * F_IN;       // B col = W row

  v8f c = {};
#pragma unroll
  for (int k0 = 0; k0 < F_IN; k0 += 4) {
    const int kk = k0 + 2 * half;
    v2f a = *(const v2f*)(arow + kk);
    v2f b = *(const v2f*)(brow + kk);
    // 8 args: (neg_a, A, neg_b, B, c_mod, C, reuse_a, reuse_b)
    c = __builtin_amdgcn_wmma_f32_16x16x4_f32(false, a, false, b,
                                              (short)0, c, false, false);
  }

  // D layout: VGPR r, lanes 0-15 -> (M=r, N=lane); lanes 16-31 -> (M=8+r, N=lane-16)
  float* yt = Y + ((long long)blockIdx.x * 16 + 8 * half) * F_OUT + wave * 16 + m;
#pragma unroll
  for (int r = 0; r < 8; ++r) {
    yt[(long long)r * F_OUT] = c[r];
  }
}

// ------------------------------------------------------------------
// edge scatter: agg[col] += x[row] * dis[row]*dis[col]
// 16 threads per edge, float4 per thread
// ------------------------------------------------------------------
__global__ void k_scatter(const int* __restrict__ row, const int* __restrict__ col,
                          const float* __restrict__ dis,
                          const float4* __restrict__ x4,
                          float* __restrict__ agg, int E) {
  long long t = (long long)blockIdx.x * blockDim.x + threadIdx.x;
  if (t >= (long long)E * 16) return;
  const int e = (int)(t >> 4);
  const int q = (int)(t & 15);
  const int r = row[e];
  const int c = col[e];
  const float norm = dis[r] * dis[c];
  float4 v = x4[(long long)r * (F_OUT / 4) + q];
  float* dst = agg + (long long)c * F_OUT + q * 4;
  atomAddF(dst + 0, v.x * norm);
  atomAddF(dst + 1, v.y * norm);
  atomAddF(dst + 2, v.z * norm);
  atomAddF(dst + 3, v.w * norm);
}

// ------------------------------------------------------------------
// per-node epilogue + pooling:
// out = relu(agg + x*dis^2 + b); sums[batch] += out; counts[batch] += 1
// 16 threads per node, float4 per thread
// ------------------------------------------------------------------
__global__ void k_pool(const float4* __restrict__ agg4, const float4* __restrict__ x4,
                       const float* __restrict__ dis, const float4* __restrict__ b4,
                       const int* __restrict__ batch,
                       float* __restrict__ sums, float* __restrict__ counts, int N) {
  long long t = (long long)blockIdx.x * blockDim.x + threadIdx.x;
  if (t >= (long long)N * 16) return;
  const int i = (int)(t >> 4);
  const int q = (int)(t & 15);
  const float d = dis[i];
  const float s = d * d;                       // self-loop norm = 1/deg
  float4 a  = agg4[(long long)i * (F_OUT / 4) + q];
  float4 xv = x4 [(long long)i * (F_OUT / 4) + q];
  float4 bv = b4[q];
  float4 o;
  o.x = fmaxf(a.x + xv.x * s + bv.x, 0.0f);
  o.y = fmaxf(a.y + xv.y * s + bv.y, 0.0f);
  o.z = fmaxf(a.z + xv.z * s + bv.z, 0.0f);
  o.w = fmaxf(a.w + xv.w * s + bv.w, 0.0f);
  const int g = batch[i];
  float* dst = sums + (long long)g * F_OUT + q * 4;
  atomAddF(dst + 0, o.x);
  atomAddF(dst + 1, o.y);
  atomAddF(dst + 2, o.z);
  atomAddF(dst + 3, o.w);
  if (q == 0) atomAddF(&counts[g], 1.0f);
}

// ------------------------------------------------------------------
// out[g,f] = sums[g,f] / max(counts[g], 1)
// ------------------------------------------------------------------
__global__ void k_final(const float* __restrict__ sums, const float* __restrict__ counts,
                        float* __restrict__ out, int G) {
  int t = blockIdx.x * blockDim.x + threadIdx.x;
  if (t < G * F_OUT) {
    out[t] = sums[t] / fmaxf(counts[t >> 6], 1.0f);
  }
}

// ------------------------------------------------------------------
extern "C" void kernel_launch(void* const* d_in, const int* in_sizes, int n_in,
                              void* d_out, int out_size, void* d_ws, size_t ws_size,
                              hipStream_t stream) {
  const float* node_feat = (const float*)d_in[0];
  const int*   edge_idx  = (const int*)d_in[1];
  const int*   batch     = (const int*)d_in[2];
  const float* W         = (const float*)d_in[3];
  const float* b         = (const float*)d_in[4];

  const int N = in_sizes[2];           // 100000
  const int E = in_sizes[1] / 2;       // 1250000
  const int G = out_size / F_OUT;      // 256

  const int* e_row = edge_idx;         // source
  const int* e_col = edge_idx + E;     // target

  // workspace layout (floats)
  float* x      = (float*)d_ws;                          // N * F_OUT
  float* agg    = x + (size_t)N * F_OUT;                 // N * F_OUT
  float* deg    = agg + (size_t)N * F_OUT;               // N   (becomes dis)
  float* sums   = deg + (size_t)N;                       // G * F_OUT
  float* counts = sums + (size_t)G * F_OUT;              // G

  const int B = 256;

  long long initN = (long long)N * F_OUT;
  k_init<<<(int)((initN + B - 1) / B), B, 0, stream>>>(agg, deg, sums, counts, N, G);

  k_degree<<<(E + B - 1) / B, B, 0, stream>>>(e_col, deg, E);

  k_rsqrt<<<(N + B - 1) / B, B, 0, stream>>>(deg, N);

  // one block per 16-row tile; 4 waves cover the 64 output features
  k_gemm_wmma<<<N / 16, 128, 0, stream>>>(node_feat, W, x);

  long long scatT = (long long)E * 16;
  k_scatter<<<(int)((scatT + B - 1) / B), B, 0, stream>>>(e_row, e_col, deg,
                                                          (const float4*)x, agg, E);

  long long poolT = (long long)N * 16;
  k_pool<<<(int)((poolT + B - 1) / B), B, 0, stream>>>((const float4*)agg,
                                                       (const float4*)x, deg,
                                                       (const float4*)b, batch,
                                                       sums, counts, N);

  k_final<<<(G * F_OUT + B - 1) / B, B, 0, stream>>>(sums, counts, (float*)d_out, G);
}